// VGG16RoIHead_32538672234527
// MI455X (gfx1250) — compile-verified
//
#include <hip/hip_runtime.h>
#include <math.h>

// CDNA5 / gfx1250, wave32.
typedef float v2f __attribute__((ext_vector_type(2)));
typedef float v8f __attribute__((ext_vector_type(8)));

#define RP_P 7
#define RP_C 512
#define RP_H 38
#define RP_W 38
#define NROI 256
#define D_IN 25088   // 512*7*7
#define D_HID 4096
#define NW 4096      // output width of both big GEMMs (compile-time!)

// ---------------------------------------------------------------------------
// Kernel 1: ROI adaptive max pool. One thread per output element (N*C*49).
// Feature map (2.9MB) is L2/WGP$-resident; negligible cost.
// ---------------------------------------------------------------------------
__global__ __launch_bounds__(256) void roi_pool_kernel(
    const float* __restrict__ fm, const float* __restrict__ rois,
    float* __restrict__ out, int total) {
  int idx = blockIdx.x * blockDim.x + threadIdx.x;
  if (idx >= total) return;
  int pw = idx % RP_P;
  int ph = (idx / RP_P) % RP_P;
  int c  = (idx / (RP_P * RP_P)) % RP_C;
  int n  = idx / (RP_P * RP_P * RP_C);

  const float* r = rois + n * 4;
  int y0 = (int)(r[0] * 0.0625f);
  int x0 = (int)(r[1] * 0.0625f);
  int y1 = (int)(r[2] * 0.0625f);
  int x1 = (int)(r[3] * 0.0625f);
  int h = y1 - y0 + 1;
  int w = x1 - x0 + 1;

  int hs = (ph * h) / RP_P;
  int hl = ((ph + 1) * h + RP_P - 1) / RP_P - hs;
  int ws_ = (pw * w) / RP_P;
  int wl = ((pw + 1) * w + RP_P - 1) / RP_P - ws_;

  const float* fmc = fm + (size_t)c * (RP_H * RP_W);
  float m = -INFINITY;
  for (int ty = 0; ty < hl; ++ty) {
    int row = y0 + hs + ty;
    row = row < 0 ? 0 : (row > RP_H - 1 ? RP_H - 1 : row);
    const float* fr = fmc + row * RP_W;
    for (int tx = 0; tx < wl; ++tx) {
      int col = x0 + ws_ + tx;
      col = col < 0 ? 0 : (col > RP_W - 1 ? RP_W - 1 : col);
      float v = fr[col];
      m = v > m ? v : m;
    }
  }
  out[(size_t)n * D_IN + c * 49 + ph * 7 + pw] = m;
}

// ---------------------------------------------------------------------------
// Kernels 2/3: fp32 WMMA GEMM  Out(256,4096) = relu(A(256,K) @ W(K,4096) + b).
//
// grid.x = 4096/32 = 128 (one 32-wide N block per workgroup).
// 8 waves/block; wave w owns M rows [32w, 32w+32) (two 16-row M tiles) for
// BOTH 16-wide N strips of the block -> 4 accumulators (2M x 2N), so every
// loaded B fragment feeds 2 WMMAs and every A fragment feeds 2 WMMAs.
// Weights stream from HBM exactly once (each block reads a distinct column
// strip); the 25.7MB activation matrix stays resident in the 192MB L2.
//
// All B addressing is immediate-offset off ONE bumped base pointer:
//   row stride 4096 floats = 16KB, second N strip = +64B, all < 24-bit ioffset.
//
// V_WMMA_F32_16X16X4_F32 fragment layouts (ISA 7.12.2, 32-bit):
//   A 16x4 : lanes 0-15 -> (M=lane, K=k0/k0+1); lanes 16-31 -> K=k0+2/k0+3
//   B 4x16 : lanes 0-15 -> (K=k0/k0+1, N=lane); lanes 16-31 -> K=k0+2/k0+3
//   C/D    : VGPR r: lanes 0-15 -> M=r, lanes 16-31 -> M=r+8
// ---------------------------------------------------------------------------
__global__ __launch_bounds__(256) void gemm_wmma_f32(
    const float* __restrict__ A, const float* __restrict__ W,
    const float* __restrict__ bias, float* __restrict__ Out, int K) {
  const int lane = threadIdx.x & 31;
  const int wave = threadIdx.x >> 5;
  const int n0   = blockIdx.x * 32;
  const int m0   = wave * 32;
  const int half = lane >> 4;     // 0: lanes 0-15, 1: lanes 16-31
  const int l15  = lane & 15;
  const int kofs = 2 * half;      // K sub-offset owned by this half-wave

  // Bumped base pointers; everything inside the unrolled body is imm-offset.
  const float* ap0 = A + (size_t)(m0 + l15) * K + kofs;
  const float* ap1 = A + (size_t)(m0 + 16 + l15) * K + kofs;
  const float* bp  = W + (size_t)kofs * NW + n0 + l15;

  v8f acc00 = {};  // M tile 0, N strip 0
  v8f acc01 = {};  // M tile 0, N strip 1
  v8f acc10 = {};  // M tile 1, N strip 0
  v8f acc11 = {};  // M tile 1, N strip 1

  for (int k0 = 0; k0 < K; k0 += 32) {
    // Prefetch weights two outer iterations ahead (global_prefetch_b8).
    __builtin_prefetch(bp + (size_t)64 * NW, 0, 1);
#pragma unroll
    for (int kk = 0; kk < 32; kk += 4) {
      v2f a0 = *(const v2f*)(ap0 + kk);
      v2f a1 = *(const v2f*)(ap1 + kk);
      const float* bb = bp + (size_t)kk * NW;   // constant -> imm offsets
      v2f b0, b1;
      b0.x = bb[0];
      b1.x = bb[16];          // +64B: second N strip
      b0.y = bb[NW];          // +16KB: row k+1
      b1.y = bb[NW + 16];
      acc00 = __builtin_amdgcn_wmma_f32_16x16x4_f32(
          false, a0, false, b0, (short)0, acc00, false, false);
      acc01 = __builtin_amdgcn_wmma_f32_16x16x4_f32(
          false, a0, false, b1, (short)0, acc01, false, false);
      acc10 = __builtin_amdgcn_wmma_f32_16x16x4_f32(
          false, a1, false, b0, (short)0, acc10, false, false);
      acc11 = __builtin_amdgcn_wmma_f32_16x16x4_f32(
          false, a1, false, b1, (short)0, acc11, false, false);
    }
    ap0 += 32;
    ap1 += 32;
    bp  += (size_t)32 * NW;
  }

  float bv0 = bias[n0 + l15];
  float bv1 = bias[n0 + 16 + l15];
  float* o0 = Out + (size_t)(m0 + 8 * half) * NW + n0 + l15;
#pragma unroll
  for (int r = 0; r < 8; ++r) {
    float v;
    v = acc00[r] + bv0; o0[(size_t)r * NW]                    = v > 0.f ? v : 0.f;
    v = acc01[r] + bv1; o0[(size_t)r * NW + 16]               = v > 0.f ? v : 0.f;
    v = acc10[r] + bv0; o0[(size_t)(r + 16) * NW]             = v > 0.f ? v : 0.f;
    v = acc11[r] + bv1; o0[(size_t)(r + 16) * NW + 16]        = v > 0.f ? v : 0.f;
  }
}

// ---------------------------------------------------------------------------
// Kernel 4: tiny heads. locs (256x84) + scores (256x21); one wave per output
// scalar, wave32 strided dot + shuffle reduction. ~220 MFLOP total.
// ---------------------------------------------------------------------------
__global__ __launch_bounds__(256) void head_kernel(
    const float* __restrict__ fc7,
    const float* __restrict__ Wloc, const float* __restrict__ bloc,
    const float* __restrict__ Wsc,  const float* __restrict__ bsc,
    float* __restrict__ out) {
  const int lane = threadIdx.x & 31;
  const int wave = threadIdx.x >> 5;
  int gw = blockIdx.x * 8 + wave;            // [0, 256*105)
  int n = gw / 105;
  int j = gw % 105;
  const float* x = fc7 + (size_t)n * D_HID;

  float s = 0.f;
  if (j < 84) {
    const float* wc = Wloc + j;
    for (int k = lane; k < D_HID; k += 32) s += x[k] * wc[(size_t)k * 84];
  } else {
    const float* wc = Wsc + (j - 84);
    for (int k = lane; k < D_HID; k += 32) s += x[k] * wc[(size_t)k * 21];
  }
#pragma unroll
  for (int off = 16; off > 0; off >>= 1) s += __shfl_xor(s, off, 32);

  if (lane == 0) {
    if (j < 84) out[n * 84 + j] = s + bloc[j];
    else        out[NROI * 84 + n * 21 + (j - 84)] = s + bsc[j - 84];
  }
}

// ---------------------------------------------------------------------------
extern "C" void kernel_launch(void* const* d_in, const int* in_sizes, int n_in,
                              void* d_out, int out_size, void* d_ws, size_t ws_size,
                              hipStream_t stream) {
  const float* fm   = (const float*)d_in[0];
  const float* rois = (const float*)d_in[1];
  const float* W1   = (const float*)d_in[2];
  const float* b1   = (const float*)d_in[3];
  const float* W2   = (const float*)d_in[4];
  const float* b2   = (const float*)d_in[5];
  const float* Wloc = (const float*)d_in[6];
  const float* bloc = (const float*)d_in[7];
  const float* Wsc  = (const float*)d_in[8];
  const float* bsc  = (const float*)d_in[9];
  float* out = (float*)d_out;

  float* pooled = (float*)d_ws;                         // 256*25088 f32
  float* fc6    = pooled + (size_t)NROI * D_IN;         // 256*4096  f32
  float* fc7    = fc6 + (size_t)NROI * D_HID;           // 256*4096  f32

  int total = NROI * RP_C * RP_P * RP_P;                // 6,422,528
  roi_pool_kernel<<<(total + 255) / 256, 256, 0, stream>>>(fm, rois, pooled, total);

  gemm_wmma_f32<<<NW / 32, 256, 0, stream>>>(pooled, W1, b1, fc6, D_IN);
  gemm_wmma_f32<<<NW / 32, 256, 0, stream>>>(fc6, W2, b2, fc7, D_HID);

  head_kernel<<<(NROI * 105) / 8, 256, 0, stream>>>(fc7, Wloc, bloc, Wsc, bsc, out);
}